// Plenoxels_49314814492917
// MI455X (gfx1250) — compile-verified
//
#include <hip/hip_runtime.h>
#include <cstdint>

// Plenoxels / NGLOD multi-LOD trilinear sampling for MI455X (gfx1250).
// Memory-bound gather kernel: no GEMM structure -> no WMMA. CDNA5 paths used:
//  - global_load_async_to_lds_b128 (ASYNCcnt) staging of the 64KB LOD0 table
//  - s_wait_asynccnt, ds_load_b128 LDS gathers, wave32 blocks
//  - TH=NT b128 gathers for the 268MB LOD4 table (protects L2 residency of LOD1-3)
//  - global_prefetch_b8 for the sequential pts stream
//  - EXEC-predicated gathers: ~13% of points are outside the cube and issue no traffic

typedef float f4 __attribute__((ext_vector_type(4)));

#define TPB 256
#define PPB 2048            // points per block (amortizes 64KB LDS staging 8:1)
#define LOD0_ENTRIES 4096   // 16^3 entries of float4 = 64KB

__device__ __forceinline__ int iclamp(int v, int lo, int hi) {
  return v < lo ? lo : (v > hi ? hi : v);
}

template <bool NT>
__device__ __forceinline__ f4 ldcb(const f4* __restrict__ p) {
  if constexpr (NT) return __builtin_nontemporal_load(p);  // th:TH_LOAD_NT
  else              return *p;
}

// One dense-grid trilinear interpolation of a float4 descriptor.
// Matches reference: pos = p01*(res-1); c0 = clamp(floor(pos),0,res-2);
// frac = pos - c0; product weights == separable lerp along z, y, x.
template <bool NT>
__device__ __forceinline__ f4 trilerp(const f4* __restrict__ cb, int res,
                                      float x01, float y01, float z01) {
  const float r1 = (float)(res - 1);
  float fx = x01 * r1, fy = y01 * r1, fz = z01 * r1;
  int cx = iclamp((int)floorf(fx), 0, res - 2);
  int cy = iclamp((int)floorf(fy), 0, res - 2);
  int cz = iclamp((int)floorf(fz), 0, res - 2);
  float tx = fx - (float)cx, ty = fy - (float)cy, tz = fz - (float)cz;
  int base = (cx * res + cy) * res + cz;   // x*res^2 + y*res + z (row-major)
  const int sx = res * res, sy = res;
  // 8 corners as 4 contiguous 32B pairs (z, z+1) -> 8 x b128 loads
  f4 v000 = ldcb<NT>(cb + base);           f4 v001 = ldcb<NT>(cb + base + 1);
  f4 v010 = ldcb<NT>(cb + base + sy);      f4 v011 = ldcb<NT>(cb + base + sy + 1);
  f4 v100 = ldcb<NT>(cb + base + sx);      f4 v101 = ldcb<NT>(cb + base + sx + 1);
  f4 v110 = ldcb<NT>(cb + base + sx + sy); f4 v111 = ldcb<NT>(cb + base + sx + sy + 1);
  f4 a = v000 + tz * (v001 - v000);
  f4 b = v010 + tz * (v011 - v010);
  f4 c = v100 + tz * (v101 - v100);
  f4 d = v110 + tz * (v111 - v110);
  f4 e = a + ty * (b - a);
  f4 f = c + ty * (d - c);
  return e + tx * (f - e);
}

__device__ __forceinline__ void wait_asynccnt0() {
#if __has_builtin(__builtin_amdgcn_s_wait_asynccnt)
  __builtin_amdgcn_s_wait_asynccnt(0);
#else
  asm volatile("s_wait_asynccnt 0" ::: "memory");
#endif
}

__global__ __launch_bounds__(TPB) void plenoxels_kernel(
    const float* __restrict__ pts,
    const f4* __restrict__ cb0, const f4* __restrict__ cb1,
    const f4* __restrict__ cb2, const f4* __restrict__ cb3,
    const f4* __restrict__ cb4,
    float* __restrict__ out, int n) {
  __shared__ f4 s_cb0[LOD0_ENTRIES];  // 64 KB: LOD0 table staged in LDS

  // --- Stage codebook_0 into LDS via CDNA5 async global->LDS DMA path ---
  // (per-lane b128 transfers tracked on ASYNCcnt; the DS pipe then serves 8
  //  of the 40 gathers per point, off the vmem pipe, and the hot 64KB table
  //  can't be thrashed out of WGP$ by the random LOD2-4 gather stream)
  for (int t = threadIdx.x; t < LOD0_ENTRIES; t += TPB) {
    unsigned lds_off = (unsigned)(uintptr_t)(&s_cb0[t]);  // low 32 bits of generic ptr = LDS byte offset
    unsigned long long ga = (unsigned long long)(uintptr_t)(cb0 + t);
    asm volatile("global_load_async_to_lds_b128 %0, %1, off"
                 :: "v"(lds_off), "v"(ga) : "memory");
  }
  wait_asynccnt0();
  __syncthreads();

  const int start = blockIdx.x * PPB;
  int end = start + PPB; if (end > n) end = n;

  for (int i = start + (int)threadIdx.x; i < end; i += TPB) {
    // Prefetch next block-stride iteration's pts cacheline (sequential stream).
    if (i + TPB < end) __builtin_prefetch(pts + 3 * (i + TPB), 0, 0);

    float px = pts[3 * i + 0] * 0.25f;   // / SCENE_SCALE
    float py = pts[3 * i + 1] * 0.25f;
    float pz = pts[3 * i + 2] * 0.25f;
    bool inside = (fabsf(px) < 0.5f) & (fabsf(py) < 0.5f) & (fabsf(pz) < 0.5f);

    float c0 = 0.0f, c1 = 0.0f, c2 = 0.0f, sg = 0.0f;
    if (inside) {   // EXEC-masked: outside lanes issue zero gather traffic
      float x01 = fminf(fmaxf(px + 0.5f, 0.0f), 1.0f);
      float y01 = fminf(fmaxf(py + 0.5f, 0.0f), 1.0f);
      float z01 = fminf(fmaxf(pz + 0.5f, 0.0f), 1.0f);

      // Largest table first: its (likely L2-miss) gathers issue earliest and
      // overlap the address math + lerps of the smaller LODs. LOD4 (268MB >
      // 192MB L2) streams with TH=NT so LOD1-3 (~38MB) stay L2-resident.
      f4 acc = trilerp<true >(cb4, 256, x01, y01, z01);
      acc    = acc + trilerp<false>(cb3, 128, x01, y01, z01);
      acc    = acc + trilerp<false>(cb2,  64, x01, y01, z01);
      acc    = acc + trilerp<false>(cb1,  32, x01, y01, z01);
      acc    = acc + trilerp<false>(s_cb0, 16, x01, y01, z01);  // LDS (ds_load_b128)

      c0 = acc.x; c1 = acc.y; c2 = acc.z;
      sg = expf(acc.w);                  // exp(-1e5) underflow case handled by sg=0
    }

    out[3 * i + 0] = c0;
    out[3 * i + 1] = c1;
    out[3 * i + 2] = c2;
    out[3 * n + i] = sg;
  }
}

extern "C" void kernel_launch(void* const* d_in, const int* in_sizes, int n_in,
                              void* d_out, int out_size, void* d_ws, size_t ws_size,
                              hipStream_t stream) {
  (void)n_in; (void)d_ws; (void)ws_size; (void)out_size;
  const float* pts = (const float*)d_in[0];
  // d_in[1] is `d` (unused by the reference)
  const f4* cb0 = (const f4*)d_in[2];
  const f4* cb1 = (const f4*)d_in[3];
  const f4* cb2 = (const f4*)d_in[4];
  const f4* cb3 = (const f4*)d_in[5];
  const f4* cb4 = (const f4*)d_in[6];
  float* out = (float*)d_out;

  int n = in_sizes[0] / 3;                 // pts is (N,3)
  int blocks = (n + PPB - 1) / PPB;        // each block handles PPB points
  plenoxels_kernel<<<blocks, TPB, 0, stream>>>(pts, cb0, cb1, cb2, cb3, cb4, out, n);
}